// GCNBlock_43920335568926
// MI455X (gfx1250) — compile-verified
//
#include <hip/hip_runtime.h>
#include <hip/hip_bf16.h>

#define NN 100000
#define DD 128
#define EE 600000
#define BN_EPS 1e-5f
#define KSTEPS 32          // DD / 4
#define NTILES (NN / 16)   // 6250
#define TILES_PER_BLOCK 10 // 6250 / 10 = 625 blocks

typedef float v2f __attribute__((ext_vector_type(2)));
typedef float v8f __attribute__((ext_vector_type(8)));

// ---------------------------------------------------------------- init: zero agg, degs, stats
__global__ void __launch_bounds__(256) gcn_init_kernel(float* __restrict__ agg,
                                                       float* __restrict__ degO,
                                                       float* __restrict__ degI,
                                                       float* __restrict__ stats) {
    int i = blockIdx.x * blockDim.x + threadIdx.x;
    if (i < NN * DD) agg[i] = 0.0f;
    if (i < NN) { degO[i] = 0.0f; degI[i] = 0.0f; }
    if (i < 2 * DD) stats[i] = 0.0f;   // colsum[128] ++ colsumsq[128]
}

// ---------------------------------------------------------------- degree counts
__global__ void __launch_bounds__(256) gcn_degree_kernel(const int* __restrict__ src,
                                                         const int* __restrict__ dst,
                                                         float* __restrict__ degO,
                                                         float* __restrict__ degI) {
    int e = blockIdx.x * blockDim.x + threadIdx.x;
    if (e < EE) {
        atomicAdd(degO + src[e], 1.0f);
        atomicAdd(degI + dst[e], 1.0f);
    }
}

// ---------------------------------------------------------------- deg -> rsqrt(max(deg,1)) in place
__global__ void __launch_bounds__(256) gcn_rs_kernel(float* __restrict__ degO,
                                                     float* __restrict__ degI) {
    int i = blockIdx.x * blockDim.x + threadIdx.x;
    if (i < NN) {
        degO[i] = rsqrtf(fmaxf(degO[i], 1.0f));
        degI[i] = rsqrtf(fmaxf(degI[i], 1.0f));
    }
}

// ---------------------------------------------------------------- edge scatter: agg[dst] += x[src]*rsO[src]
// one wave per edge; lane handles 4 consecutive floats (32*4 = 128 = DD)
__global__ void __launch_bounds__(256) gcn_scatter_kernel(const int* __restrict__ src,
                                                          const int* __restrict__ dst,
                                                          const float* __restrict__ x,
                                                          const float* __restrict__ rsO,
                                                          float* __restrict__ agg) {
    int wave = (blockIdx.x * blockDim.x + threadIdx.x) >> 5;
    int lane = threadIdx.x & 31;
    if (wave >= EE) return;
    int s = src[wave];
    int d = dst[wave];
    float sc = rsO[s];
    const float4* xp = (const float4*)(x + (size_t)s * DD);
    float4 v = xp[lane];
    float* ap = agg + (size_t)d * DD + lane * 4;
    atomicAdd(ap + 0, v.x * sc);
    atomicAdd(ap + 1, v.y * sc);
    atomicAdd(ap + 2, v.z * sc);
    atomicAdd(ap + 3, v.w * sc);
}

// ---------------------------------------------------------------- WMMA GEMM + bias + residual + BN stats
// t = x + (diag(rs_in)*agg) @ W + b     -> written to out
// per-column sum/sumsq accumulated in registers, one atomic per lane at end.
__global__ void __launch_bounds__(256) gcn_gemm_kernel(const float* __restrict__ agg,
                                                       const float* __restrict__ rsI,
                                                       const float* __restrict__ x,
                                                       const float* __restrict__ W,
                                                       const float* __restrict__ bias,
                                                       float* __restrict__ out,
                                                       float* __restrict__ colsum,
                                                       float* __restrict__ colsumsq) {
    const int lane = threadIdx.x & 31;
    const int wid  = threadIdx.x >> 5;   // 0..7 -> N tile
    const int n0   = wid * 16;
    const int col  = lane & 15;          // N within tile
    const int kg   = lane >> 4;          // K group: 0 or 1

    // B fragments for the whole K loop (ISA 32-bit 4x16 B layout):
    // VGPR j of step ks holds W[4*ks + j + 2*kg][n0+col]
    v2f bfrag[KSTEPS];
#pragma unroll
    for (int ks = 0; ks < KSTEPS; ++ks) {
        int kr = 4 * ks + 2 * kg;
        bfrag[ks].x = W[(kr + 0) * DD + n0 + col];
        bfrag[ks].y = W[(kr + 1) * DD + n0 + col];
    }
    float bn = bias[n0 + col];

    float s1 = 0.0f, s2 = 0.0f;

    int t0 = blockIdx.x * TILES_PER_BLOCK;
    for (int t = t0; t < t0 + TILES_PER_BLOCK && t < NTILES; ++t) {
        int m0 = t * 16;
        int arow = m0 + col;             // A row for this lane
        float rs = rsI[arow];
        const float* aptr = agg + (size_t)arow * DD + 2 * kg;

        v8f acc = {};
#pragma unroll
        for (int ks = 0; ks < KSTEPS; ++ks) {
            v2f a = *(const v2f*)(aptr + 4 * ks);
            a.x *= rs;
            a.y *= rs;
            // D = A(16x4,f32) * B(4x16,f32) + C  -> v_wmma_f32_16x16x4_f32
            acc = __builtin_amdgcn_wmma_f32_16x16x4_f32(
                false, a, false, bfrag[ks], (short)0, acc, false, false);
        }

        // C layout: VGPR v -> row m0 + v + 8*kg, column n0+col
#pragma unroll
        for (int v = 0; v < 8; ++v) {
            int row = m0 + v + 8 * kg;
            size_t idx = (size_t)row * DD + n0 + col;
            float val = acc[v] + bn + x[idx];
            out[idx] = val;
            s1 += val;
            s2 += val * val;
        }
    }
    atomicAdd(colsum   + n0 + col, s1);
    atomicAdd(colsumsq + n0 + col, s2);
}

// ---------------------------------------------------------------- finalize BN coefficients (128 threads)
__global__ void gcn_bnfin_kernel(const float* __restrict__ colsum,
                                 const float* __restrict__ colsumsq,
                                 const float* __restrict__ gamma,
                                 const float* __restrict__ beta,
                                 float* __restrict__ acoef,
                                 float* __restrict__ ccoef) {
    int n = threadIdx.x;
    if (n < DD) {
        float invN = 1.0f / (float)NN;
        float mean = colsum[n] * invN;
        float var  = colsumsq[n] * invN - mean * mean;
        float inv  = rsqrtf(var + BN_EPS);
        float a = gamma[n] * inv;
        acoef[n] = a;
        ccoef[n] = beta[n] - mean * a;
    }
}

// ---------------------------------------------------------------- normalize + ReLU in place (float4)
__global__ void __launch_bounds__(256) gcn_norm_kernel(float* __restrict__ out,
                                                       const float* __restrict__ acoef,
                                                       const float* __restrict__ ccoef) {
    int i = blockIdx.x * blockDim.x + threadIdx.x;   // over NN*DD/4
    if (i >= NN * DD / 4) return;
    int nb = (i * 4) & (DD - 1);
    float4 v  = ((float4*)out)[i];
    float4 av = *(const float4*)(acoef + nb);
    float4 cv = *(const float4*)(ccoef + nb);
    v.x = fmaxf(v.x * av.x + cv.x, 0.0f);
    v.y = fmaxf(v.y * av.y + cv.y, 0.0f);
    v.z = fmaxf(v.z * av.z + cv.z, 0.0f);
    v.w = fmaxf(v.w * av.w + cv.w, 0.0f);
    ((float4*)out)[i] = v;
}

// ---------------------------------------------------------------- host launcher
extern "C" void kernel_launch(void* const* d_in, const int* in_sizes, int n_in,
                              void* d_out, int out_size, void* d_ws, size_t ws_size,
                              hipStream_t stream) {
    const float* x     = (const float*)d_in[0];
    const int*   eidx  = (const int*)d_in[1];      // [2, E]
    const float* W     = (const float*)d_in[2];
    const float* b     = (const float*)d_in[3];
    const float* gamma = (const float*)d_in[4];
    const float* beta  = (const float*)d_in[5];
    const int* src = eidx;
    const int* dst = eidx + EE;

    // workspace layout (floats)
    float* ws   = (float*)d_ws;
    float* agg  = ws;                       // N*D
    float* degO = agg + (size_t)NN * DD;    // N   (becomes rs_out)
    float* degI = degO + NN;                // N   (becomes rs_in)
    float* colsum   = degI + NN;            // 128
    float* colsumsq = colsum + DD;          // 128
    float* acoef    = colsumsq + DD;        // 128
    float* ccoef    = acoef + DD;           // 128

    float* out = (float*)d_out;

    // 1) zero
    gcn_init_kernel<<<(NN * DD + 255) / 256, 256, 0, stream>>>(agg, degO, degI, colsum);
    // 2) degrees
    gcn_degree_kernel<<<(EE + 255) / 256, 256, 0, stream>>>(src, dst, degO, degI);
    // 3) rsqrt(max(deg,1))
    gcn_rs_kernel<<<(NN + 255) / 256, 256, 0, stream>>>(degO, degI);
    // 4) edge scatter (one wave per edge -> 8 edges per 256-thread block)
    gcn_scatter_kernel<<<(EE + 7) / 8, 256, 0, stream>>>(src, dst, x, degO, agg);
    // 5) WMMA GEMM + bias + residual + stats
    gcn_gemm_kernel<<<(NTILES + TILES_PER_BLOCK - 1) / TILES_PER_BLOCK, 256, 0, stream>>>(
        agg, degI, x, W, b, out, colsum, colsumsq);
    // 6) BN coefficients
    gcn_bnfin_kernel<<<1, DD, 0, stream>>>(colsum, colsumsq, gamma, beta, acoef, ccoef);
    // 7) normalize + ReLU in place
    gcn_norm_kernel<<<(NN * DD / 4 + 255) / 256, 256, 0, stream>>>(out, acoef, ccoef);
}